// SkeletonBase_13245679141272
// MI455X (gfx1250) — compile-verified
//
#include <hip/hip_runtime.h>
#include <cstdint>

#define JOINTS     32
#define TPB        96                  // batch elements per block == floats per element (3 waves)
#define PERB       (JOINTS * 3)        // 96 floats per batch element
#define LDS_STRIDE 100                 // 4-aligned (b128-capable), mild 2-way bank conflict
#define NCHUNK     (PERB / 4)          // 24 float4 chunks per column

// ---- address-space helpers for the async-DMA builtins ----
typedef int v4i __attribute__((ext_vector_type(4)));
typedef const __attribute__((address_space(1))) void* as1_cvptr;
typedef __attribute__((address_space(1))) int*        as1_iptr;
typedef __attribute__((address_space(1))) v4i*        as1_v4iptr;
typedef __attribute__((address_space(3))) void*       as3_vptr;
typedef __attribute__((address_space(3))) int*        as3_iptr;
typedef __attribute__((address_space(3))) v4i*        as3_v4iptr;

#if defined(__HIP_DEVICE_COMPILE__)
#if __has_builtin(__builtin_amdgcn_global_load_async_to_lds_b128)
#define FK_LD_B128 1
#elif __has_builtin(__builtin_amdgcn_global_load_async_to_lds_b32)
#define FK_LD_B32 1
#endif
#if __has_builtin(__builtin_amdgcn_global_store_async_from_lds_b128)
#define FK_ST_B128 1
#elif __has_builtin(__builtin_amdgcn_global_store_async_from_lds_b32)
#define FK_ST_B32 1
#endif
#endif

__device__ __forceinline__ void wait_async0() {
#if defined(__HIP_DEVICE_COMPILE__)
#if __has_builtin(__builtin_amdgcn_s_wait_asynccnt)
    __builtin_amdgcn_s_wait_asynccnt(0);
#else
    asm volatile("s_wait_asynccnt 0" ::: "memory");
#endif
#endif
}

// R = I + sin(th)/(th+eps) * S(a) + (1-cos(th))/(th+eps)^2 * S(a)^2
// == reference's I + sin*S(r) + (1-cos)*S(r)^2 with r = a/(th+eps).
__device__ __forceinline__ void rodrigues(float ax, float ay, float az, float R[9]) {
    const float eps = 1.1920929e-07f;                 // finfo(float32).eps
    float xx = ax * ax, yy = ay * ay, zz = az * az;
    float th  = __builtin_amdgcn_sqrtf(xx + yy + zz); // v_sqrt_f32
    float inv = __builtin_amdgcn_rcpf(th + eps);      // v_rcp_f32
    float s, c;
    __sincosf(th, &s, &c);                            // v_sin_f32 / v_cos_f32
    float A = s * inv;
    float B = (1.0f - c) * (inv * inv);
    float xy = ax * ay, xz = ax * az, yz = ay * az;
    float sx = A * ax, sy = A * ay, sz = A * az;
    R[0] = 1.0f - B * (yy + zz);  R[1] = B * xy - sz;           R[2] = B * xz + sy;
    R[3] = B * xy + sz;           R[4] = 1.0f - B * (xx + zz);  R[5] = B * yz - sx;
    R[6] = B * xz - sy;           R[7] = B * yz + sx;           R[8] = 1.0f - B * (xx + yy);
}

__global__ __launch_bounds__(TPB) void fk_fused(
    const float* __restrict__ angles,    // [B, 32, 3]
    const float* __restrict__ hip_pos,   // [B, 3]
    const float* __restrict__ offsets,   // [32, 3]  (uniform -> s_load)
    float* __restrict__ out,             // [B, 32, 3]
    int batch)
{
    __shared__ float lds[TPB * LDS_STRIDE];           // 38.4 KB
    const int t = threadIdx.x;
    const long long firstB = (long long)blockIdx.x * TPB;

    // one 64-bit base add per buffer; everything else is 32-bit affine
    const float* gsrc    = angles  + firstB * PERB;
    float*       gdst    = out     + firstB * PERB;
    const float* hipBase = hip_pos + firstB * 3;

    long long rem64 = (long long)batch - firstB;
    const int remB  = rem64 > TPB ? TPB : (int)rem64; // batch elems in this block
    const bool full = (remB == TPB);

    // float4-chunk decomposition: chunk Q = it*96 + t -> element e = 4*it + a, slot r
    const int a = t / 24;               // per-thread constant (0..3)
    const int r = t - 24 * a;           // per-thread constant (0..23)
    const int ldsB128 = 100 * a + 4 * r;

    // ---------- Phase 1: angles global -> LDS via async DMA (coalesced) ----------
#if defined(FK_LD_B128)
    if (full) {
#pragma unroll
        for (int it = 0; it < NCHUNK; ++it)
            __builtin_amdgcn_global_load_async_to_lds_b128(
                (as1_v4iptr)(as1_cvptr)(gsrc + 384 * it + 4 * t),
                (as3_v4iptr)(as3_vptr)&lds[400 * it + ldsB128],
                0, 0);
    } else {
        for (int it = 0; it < NCHUNK; ++it)
            if (4 * it + a < remB)
                __builtin_amdgcn_global_load_async_to_lds_b128(
                    (as1_v4iptr)(as1_cvptr)(gsrc + 384 * it + 4 * t),
                    (as3_v4iptr)(as3_vptr)&lds[400 * it + ldsB128],
                    0, 0);
    }
    wait_async0();
#elif defined(FK_LD_B32)
    {
        const int n = full ? TPB : remB;              // iteration it copies element it
        for (int it = 0; it < n; ++it)
            __builtin_amdgcn_global_load_async_to_lds_b32(
                (as1_iptr)(as1_cvptr)(gsrc + PERB * it + t),
                (as3_iptr)(as3_vptr)&lds[LDS_STRIDE * it + t],
                0, 0);
    }
    wait_async0();
#else
    for (int it = 0; it < NCHUNK; ++it) {
        if (4 * it + a < remB) {
            const float4 v = *reinterpret_cast<const float4*>(gsrc + 384 * it + 4 * t);
            float* d = &lds[400 * it + ldsB128];
            d[0] = v.x; d[1] = v.y; d[2] = v.z; d[3] = v.w;
        }
    }
#endif
    __syncthreads();

    // ---------- Phase 2: per-lane kinematic chain ----------
    if (t < remB) {
        float* col = &lds[t * LDS_STRIDE];            // this lane's 96-float column
        static constexpr int PAR[JOINTS] = {
            -1, 0, 1, 2, 3, 4, 0, 6, 7, 8, 9, 0, 11, 12, 13, 14,
            12, 16, 17, 18, 19, 20, 19, 22, 12, 24, 25, 26, 27, 28, 27, 30};

        float Rg[JOINTS][9];                          // constant-indexed -> registers
        float P [JOINTS][3];

        rodrigues(col[0], col[1], col[2], Rg[0]);
        P[0][0] = hipBase[3 * t + 0];
        P[0][1] = hipBase[3 * t + 1];
        P[0][2] = hipBase[3 * t + 2];
        col[0] = P[0][0]; col[1] = P[0][1]; col[2] = P[0][2];

#pragma unroll
        for (int j = 1; j < JOINTS; ++j) {
            const int p = PAR[j];
            float Rl[9];
            rodrigues(col[3 * j + 0], col[3 * j + 1], col[3 * j + 2], Rl);

            const float ox = offsets[3 * j + 0];      // uniform scalars (s_load)
            const float oy = offsets[3 * j + 1];
            const float oz = offsets[3 * j + 2];
#pragma unroll
            for (int m = 0; m < 3; ++m) {
                // pos_j = offsets[j] (row) @ Rg[parent] + pos[parent]
                P[j][m] = fmaf(ox, Rg[p][0 + m],
                          fmaf(oy, Rg[p][3 + m],
                          fmaf(oz, Rg[p][6 + m], P[p][m])));
            }
#pragma unroll
            for (int i = 0; i < 3; ++i)
#pragma unroll
                for (int m = 0; m < 3; ++m)
                    Rg[j][3 * i + m] = fmaf(Rl[3 * i + 0], Rg[p][0 + m],
                                       fmaf(Rl[3 * i + 1], Rg[p][3 + m],
                                            Rl[3 * i + 2] * Rg[p][6 + m]));
            col[3 * j + 0] = P[j][0];
            col[3 * j + 1] = P[j][1];
            col[3 * j + 2] = P[j][2];
        }
    }
    __syncthreads();

    // ---------- Phase 3: LDS -> global via async DMA stores (coalesced) ----------
#if defined(FK_ST_B128)
    if (full) {
#pragma unroll
        for (int it = 0; it < NCHUNK; ++it)
            __builtin_amdgcn_global_store_async_from_lds_b128(
                (as1_v4iptr)(as1_cvptr)(gdst + 384 * it + 4 * t),
                (as3_v4iptr)(as3_vptr)&lds[400 * it + ldsB128],
                0, 0);
    } else {
        for (int it = 0; it < NCHUNK; ++it)
            if (4 * it + a < remB)
                __builtin_amdgcn_global_store_async_from_lds_b128(
                    (as1_v4iptr)(as1_cvptr)(gdst + 384 * it + 4 * t),
                    (as3_v4iptr)(as3_vptr)&lds[400 * it + ldsB128],
                    0, 0);
    }
    wait_async0();   // S_ENDPGM also does implicit wait-idle; belt-and-braces
#elif defined(FK_ST_B32)
    {
        const int n = full ? TPB : remB;
        for (int it = 0; it < n; ++it)
            __builtin_amdgcn_global_store_async_from_lds_b32(
                (as1_iptr)(as1_cvptr)(gdst + PERB * it + t),
                (as3_iptr)(as3_vptr)&lds[LDS_STRIDE * it + t],
                0, 0);
    }
    wait_async0();
#else
    for (int it = 0; it < NCHUNK; ++it) {
        if (4 * it + a < remB) {
            const float* s = &lds[400 * it + ldsB128];
            float4 v;
            v.x = s[0]; v.y = s[1]; v.z = s[2]; v.w = s[3];
            *reinterpret_cast<float4*>(gdst + 384 * it + 4 * t) = v;
        }
    }
#endif
}

extern "C" void kernel_launch(void* const* d_in, const int* in_sizes, int n_in,
                              void* d_out, int out_size, void* d_ws, size_t ws_size,
                              hipStream_t stream) {
    const float* angles  = (const float*)d_in[0];   // [B,32,3]
    const float* hip_pos = (const float*)d_in[1];   // [B,3]
    const float* offsets = (const float*)d_in[2];   // [32,3]
    float* out = (float*)d_out;

    const int batch  = in_sizes[0] / PERB;          // 131072
    const int blocks = (batch + TPB - 1) / TPB;     // 1366
    fk_fused<<<blocks, TPB, 0, stream>>>(angles, hip_pos, offsets, out, batch);
}